// DiffLogicCA_54674933678727
// MI455X (gfx1250) — compile-verified
//
#include <hip/hip_runtime.h>

typedef __attribute__((ext_vector_type(2))) float v2f;
typedef __attribute__((ext_vector_type(8))) float v8f;

#define NGATES 2415

// GATE_COEFFS from the reference (16 x 4)
__device__ const float G_TAB[16][4] = {
    {0, 0, 0, 0},   {0, 0, 0, 1},   {0, 1, 0, -1},  {0, 1, 0, 0},
    {0, 0, 1, -1},  {0, 0, 1, 0},   {0, 1, 1, -2},  {0, 1, 1, -1},
    {1, -1, -1, 1}, {1, -1, -1, 2}, {1, 0, -1, 0},  {1, 0, -1, 1},
    {1, -1, 0, 0},  {1, -1, 0, 1},  {1, 0, 0, -1},  {1, 0, 0, 0}};

// ---------------------------------------------------------------------------
// Precompute: per-gate coef[4] = softmax(w) @ G, plus packed (ia,ib) indices.
// One wave handles 16 gates. The 16x4 projection is done with chained
// V_WMMA_F32_16X16X4_F32 (M=16 gates, K=16 probs in 4 steps, N=4 used cols).
// f32 A layout: lanes 0-15 = M, VGPR0 holds K=k0 (lo lanes) / K=k0+2 (hi),
// VGPR1 holds K=k0+1 / k0+3.  C/D: VGPR r = row r (lo lanes) / r+8 (hi).
// ---------------------------------------------------------------------------
__global__ __launch_bounds__(32) void coef_kernel(
    const float* __restrict__ w, const int* __restrict__ ia,
    const int* __restrict__ ib, float* __restrict__ coefOut,
    int* __restrict__ pidx, int n, int goff) {
  const int lane = threadIdx.x;
  const int m = lane & 15;   // matrix row for A / column N for B,C,D
  const int hi = lane >> 4;  // lane-half selects K (A,B) / M+8 (C,D)
  const int gbase = blockIdx.x * 16;
  const int g = gbase + m;
  const bool valid = g < n;

  float p[16];
  float mx = -3.402823466e38f;
#pragma unroll
  for (int i = 0; i < 16; ++i) {
    p[i] = valid ? w[g * 16 + i] : 0.0f;
    mx = fmaxf(mx, p[i]);
  }
  float s = 0.0f;
#pragma unroll
  for (int i = 0; i < 16; ++i) {
    p[i] = expf(p[i] - mx);
    s += p[i];
  }
  const float inv = 1.0f / s;
#pragma unroll
  for (int i = 0; i < 16; ++i) p[i] = valid ? p[i] * inv : 0.0f;

#if __has_builtin(__builtin_amdgcn_wmma_f32_16x16x4_f32)
  v8f acc = {0.f, 0.f, 0.f, 0.f, 0.f, 0.f, 0.f, 0.f};
#pragma unroll
  for (int k0 = 0; k0 < 16; k0 += 4) {
    v2f A, Bv;
    A.x = hi ? p[k0 + 2] : p[k0 + 0];
    A.y = hi ? p[k0 + 3] : p[k0 + 1];
    const int ra = k0 + 2 * hi;
    Bv.x = (m < 4) ? G_TAB[ra + 0][m] : 0.0f;
    Bv.y = (m < 4) ? G_TAB[ra + 1][m] : 0.0f;
    acc = __builtin_amdgcn_wmma_f32_16x16x4_f32(false, A, false, Bv, (short)0,
                                                acc, false, false);
  }
  if (m < 4) {
#pragma unroll
    for (int r = 0; r < 8; ++r) {
      const int gg = gbase + r + 8 * hi;  // row M=r (lo) / r+8 (hi), col N=m
      if (gg < n) coefOut[(goff + gg) * 4 + m] = acc[r];
    }
  }
#else
  if (valid && hi == 0) {
    float c0 = 0.f, c1 = 0.f, c2 = 0.f, c3 = 0.f;
#pragma unroll
    for (int i = 0; i < 16; ++i) {
      c0 += p[i] * G_TAB[i][0];
      c1 += p[i] * G_TAB[i][1];
      c2 += p[i] * G_TAB[i][2];
      c3 += p[i] * G_TAB[i][3];
    }
    coefOut[(goff + g) * 4 + 0] = c0;
    coefOut[(goff + g) * 4 + 1] = c1;
    coefOut[(goff + g) * 4 + 2] = c2;
    coefOut[(goff + g) * 4 + 3] = c3;
  }
#endif
  if (valid && hi == 0)
    pidx[goff + g] = (ia[g] & 0xFFFF) | (ib[g] << 16);
}

// ---------------------------------------------------------------------------
// Main kernel: 1 wave per block, 2 cells per lane (float2), LDS column per
// lane with stride 32 float2 -> ds_load_b64 hits all 64 banks, no conflicts,
// no barriers. Tables (coef/pidx) are wave-uniform loads.
// ---------------------------------------------------------------------------
__device__ __forceinline__ float2 evalGate(int g, const float2* col, int srcOff,
                                           const float4* __restrict__ coef,
                                           const int* __restrict__ pidx) {
  const unsigned id = (unsigned)pidx[g];
  const int ea = (int)(id & 0xFFFFu);
  const int eb = (int)(id >> 16);
  const float4 c = coef[g];
  const float2 a = col[(srcOff + ea) * 32];
  const float2 b = col[(srcOff + eb) * 32];
  float2 r;
  r.x = fmaf(a.x, fmaf(c.w, b.x, c.y), fmaf(c.z, b.x, c.x));
  r.y = fmaf(a.y, fmaf(c.w, b.y, c.y), fmaf(c.z, b.y, c.x));
  return r;
}

template <int O>
__device__ __forceinline__ void updLayer(int goff, float2* col, int s, int d,
                                         const float4* __restrict__ coef,
                                         const int* __restrict__ pidx) {
#pragma unroll 8
  for (int j = 0; j < O; ++j)
    col[(d + j) * 32] = evalGate(goff + j, col, s, coef, pidx);
}

__global__ __launch_bounds__(32) void ca_main(const float* __restrict__ grid,
                                              const float4* __restrict__ coef,
                                              const int* __restrict__ pidx,
                                              float* __restrict__ out,
                                              int B) {
  __shared__ float2 ubuf[256][32];  // 64 KB: entries 0..127 / 128..255 ping
  const int tid = threadIdx.x;
  float2* col = &ubuf[0][tid];
  const int c0 = (blockIdx.x * 32 + tid) * 2;
  if (c0 >= B) return;

  // x: 9 values for two cells, packed as float2, staged once into entries 0..8
#pragma unroll
  for (int e = 0; e < 9; ++e)
    col[e * 32] = make_float2(grid[c0 * 9 + e], grid[c0 * 9 + 9 + e]);

  // ---- perception: 16 channels, widths 9->8->4->2->1 ----
  float2 P[16];
#pragma unroll 1
  for (int k = 0; k < 16; ++k) {
#pragma unroll
    for (int j = 0; j < 8; ++j)  // L0: src entries 0..8 (x), dst 16..23
      col[(16 + j) * 32] = evalGate(k * 8 + j, col, 0, coef, pidx);
#pragma unroll
    for (int j = 0; j < 4; ++j)  // L1: src 16..23, dst 24..27
      col[(24 + j) * 32] = evalGate(128 + k * 4 + j, col, 16, coef, pidx);
#pragma unroll
    for (int j = 0; j < 2; ++j)  // L2: src 24..27, dst 16..17
      col[(16 + j) * 32] = evalGate(192 + k * 2 + j, col, 24, coef, pidx);
    P[k] = evalGate(224 + k, col, 16, coef, pidx);  // L3 -> register
  }

  // u = concat(x, perception): entries 0..8 still hold x, add 9..24
#pragma unroll
  for (int k = 0; k < 16; ++k) col[(9 + k) * 32] = P[k];

  // ---- update: 16 x 128-wide, then 64,32,16,8,4,2,1 ----
  int s = 0, d = 128, goff = 240;
#pragma unroll 1
  for (int l = 0; l < 16; ++l) {
    updLayer<128>(goff, col, s, d, coef, pidx);
    goff += 128; s ^= 128; d ^= 128;
  }
  updLayer<64>(goff, col, s, d, coef, pidx); goff += 64; s ^= 128; d ^= 128;
  updLayer<32>(goff, col, s, d, coef, pidx); goff += 32; s ^= 128; d ^= 128;
  updLayer<16>(goff, col, s, d, coef, pidx); goff += 16; s ^= 128; d ^= 128;
  updLayer<8>(goff, col, s, d, coef, pidx);  goff += 8;  s ^= 128; d ^= 128;
  updLayer<4>(goff, col, s, d, coef, pidx);  goff += 4;  s ^= 128; d ^= 128;
  updLayer<2>(goff, col, s, d, coef, pidx);  goff += 2;  s ^= 128; d ^= 128;
  const float2 r = evalGate(goff, col, s, coef, pidx);  // final gate, o=1

  out[c0] = r.x;
  out[c0 + 1] = r.y;
}

// ---------------------------------------------------------------------------
// Host
// ---------------------------------------------------------------------------
extern "C" void kernel_launch(void* const* d_in, const int* in_sizes, int n_in,
                              void* d_out, int out_size, void* d_ws,
                              size_t ws_size, hipStream_t stream) {
  (void)n_in; (void)out_size; (void)ws_size;
  const float* grid = (const float*)d_in[0];
  // input order: grid, perc_ia[4], perc_ib[4], perc_w[4],
  //              upd_ia[23], upd_ib[23], upd_w[23]
  float4* coef = (float4*)d_ws;
  int* pidx = (int*)((char*)d_ws + (size_t)NGATES * sizeof(float4));

  static const int PERC_OUT[4] = {8, 4, 2, 1};
  static const int UPD_OUT[23] = {128, 128, 128, 128, 128, 128, 128, 128,
                                  128, 128, 128, 128, 128, 128, 128, 128,
                                  64,  32,  16,  8,   4,   2,   1};

  int goff = 0;
  for (int l = 0; l < 4; ++l) {
    const int n = 16 * PERC_OUT[l];  // gates laid out as (K, o) row-major
    coef_kernel<<<(n + 15) / 16, 32, 0, stream>>>(
        (const float*)d_in[9 + l], (const int*)d_in[1 + l],
        (const int*)d_in[5 + l], (float*)coef, pidx, n, goff);
    goff += n;
  }
  for (int l = 0; l < 23; ++l) {
    const int n = UPD_OUT[l];
    coef_kernel<<<(n + 15) / 16, 32, 0, stream>>>(
        (const float*)d_in[59 + l], (const int*)d_in[13 + l],
        (const int*)d_in[36 + l], (float*)coef, pidx, n, goff);
    goff += n;
  }

  const int B = in_sizes[0] / 9;  // 131072
  const int nblk = (B / 2 + 31) / 32;
  ca_main<<<nblk, 32, 0, stream>>>(grid, coef, pidx, (float*)d_out, B);
}